// Mamba_40346922778706
// MI455X (gfx1250) — compile-verified
//
#include <hip/hip_runtime.h>
#include <math.h>
#include <stdint.h>

#define B_  4
#define L_  2048
#define DM  1024
#define DS  16
#define DC  4
#define DI  2048      // EXP * DM
#define DR  64        // (DM + 15) / 16
#define BL  (B_ * L_) // 8192

typedef float v2f __attribute__((ext_vector_type(2)));
typedef float v8f __attribute__((ext_vector_type(8)));
typedef int   v4i __attribute__((ext_vector_type(4)));

#if __has_builtin(__builtin_amdgcn_global_load_async_to_lds_b128)
#define USE_ASYNC_LDS 1
typedef __attribute__((address_space(1))) v4i as1_v4i;   // global
typedef __attribute__((address_space(3))) v4i as3_v4i;   // LDS
#endif

#ifdef USE_ASYNC_LDS
__device__ __forceinline__ void wait_async_all() {
#if __has_builtin(__builtin_amdgcn_s_wait_asynccnt)
    __builtin_amdgcn_s_wait_asynccnt(0);
#else
    asm volatile("s_wait_asynccnt 0" ::: "memory");
#endif
}
#endif

__device__ __forceinline__ float sigmoid_fast(float x) {
    return 1.0f / (1.0f + __expf(-x));
}
__device__ __forceinline__ float softplus_fast(float x) {
    // numerically stable: max(x,0) + log1p(exp(-|x|))
    return fmaxf(x, 0.0f) + log1pf(__expf(-fabsf(x)));
}

// ---------------------------------------------------------------------------
// C[M,N] = A[M,K] (row stride lda) @ W[N,K]^T, fp32 via V_WMMA_F32_16X16X4_F32.
// One wave -> 16(M) x 64(N) strip: one A fragment reused by 4 B fragments.
// K is consumed in blocks of 16 (4 WMMA k-steps) with register double
// buffering: the next block's 20 global_load_b64 are issued before the
// current block's 16 WMMAs, so waits become partial (latency hiding).
// EPI==1: C = softplus(C + bias[n]).
// Fragment layout (ISA 7.12.2, 32-bit 16x4): lane l holds row/col (l&15),
// K pair 2*(l>>4) -> contiguous 8B loads from row-major A and row-major W.
// ---------------------------------------------------------------------------
template <int EPI>
__global__ __launch_bounds__(256) void gemm_xwt_kernel(
    const float* __restrict__ A, int lda,
    const float* __restrict__ W,
    const float* __restrict__ bias,
    float* __restrict__ C,
    int M, int N, int K)   // K must be a multiple of 16
{
    const int lane    = threadIdx.x & 31;
    const int wave    = blockIdx.x * (blockDim.x >> 5) + (threadIdx.x >> 5);
    const int strips  = (N + 63) >> 6;
    const int m_tile  = wave / strips;
    const int n_strip = wave % strips;
    if (m_tile * 16 >= M) return;               // wave-uniform: EXEC stays all-1

    const int m0 = m_tile * 16;
    const int n0 = n_strip * 64;
    const int r  = lane & 15;
    const int kh = (lane >> 4) << 1;            // 0 or 2

    const float* arow = A + (size_t)(m0 + r) * lda + kh;
    const int n0r = (n0 +  0 + r) < N ? (n0 +  0 + r) : 0;
    const int n1r = (n0 + 16 + r) < N ? (n0 + 16 + r) : 0;
    const int n2r = (n0 + 32 + r) < N ? (n0 + 32 + r) : 0;
    const int n3r = (n0 + 48 + r) < N ? (n0 + 48 + r) : 0;
    const float* w0 = W + (size_t)n0r * K + kh;
    const float* w1 = W + (size_t)n1r * K + kh;
    const float* w2 = W + (size_t)n2r * K + kh;
    const float* w3 = W + (size_t)n3r * K + kh;

    v8f acc0 = {}; v8f acc1 = {}; v8f acc2 = {}; v8f acc3 = {};

    v2f a[4], b0[4], b1[4], b2[4], b3[4];
    #pragma unroll
    for (int j = 0; j < 4; ++j) {
        a[j]  = *(const v2f*)(arow + 4 * j);
        b0[j] = *(const v2f*)(w0 + 4 * j);
        b1[j] = *(const v2f*)(w1 + 4 * j);
        b2[j] = *(const v2f*)(w2 + 4 * j);
        b3[j] = *(const v2f*)(w3 + 4 * j);
    }

    for (int k = 16; k <= K - 16; k += 16) {
        v2f an[4], bn0[4], bn1[4], bn2[4], bn3[4];
        #pragma unroll
        for (int j = 0; j < 4; ++j) {           // prefetch next K block
            an[j]  = *(const v2f*)(arow + k + 4 * j);
            bn0[j] = *(const v2f*)(w0 + k + 4 * j);
            bn1[j] = *(const v2f*)(w1 + k + 4 * j);
            bn2[j] = *(const v2f*)(w2 + k + 4 * j);
            bn3[j] = *(const v2f*)(w3 + k + 4 * j);
        }
        #pragma unroll
        for (int j = 0; j < 4; ++j) {           // consume current K block
            acc0 = __builtin_amdgcn_wmma_f32_16x16x4_f32(false, a[j], false, b0[j], (short)0, acc0, false, false);
            acc1 = __builtin_amdgcn_wmma_f32_16x16x4_f32(false, a[j], false, b1[j], (short)0, acc1, false, false);
            acc2 = __builtin_amdgcn_wmma_f32_16x16x4_f32(false, a[j], false, b2[j], (short)0, acc2, false, false);
            acc3 = __builtin_amdgcn_wmma_f32_16x16x4_f32(false, a[j], false, b3[j], (short)0, acc3, false, false);
        }
        #pragma unroll
        for (int j = 0; j < 4; ++j) {
            a[j] = an[j]; b0[j] = bn0[j]; b1[j] = bn1[j]; b2[j] = bn2[j]; b3[j] = bn3[j];
        }
    }
    #pragma unroll
    for (int j = 0; j < 4; ++j) {               // final K block
        acc0 = __builtin_amdgcn_wmma_f32_16x16x4_f32(false, a[j], false, b0[j], (short)0, acc0, false, false);
        acc1 = __builtin_amdgcn_wmma_f32_16x16x4_f32(false, a[j], false, b1[j], (short)0, acc1, false, false);
        acc2 = __builtin_amdgcn_wmma_f32_16x16x4_f32(false, a[j], false, b2[j], (short)0, acc2, false, false);
        acc3 = __builtin_amdgcn_wmma_f32_16x16x4_f32(false, a[j], false, b3[j], (short)0, acc3, false, false);
    }

    // C/D layout: VGPR i -> row m0 + i + 8*(lane>>4), col n_tile + (lane&15)
    const int mb = m0 + ((lane >> 4) << 3);
    v8f accs[4] = {acc0, acc1, acc2, acc3};
    #pragma unroll
    for (int t = 0; t < 4; ++t) {
        const int nt = n0 + t * 16;
        if (nt >= N) break;                     // wave-uniform
        const int nn = nt + r;
        const float bv = (EPI == 1) ? bias[nn] : 0.0f;
        #pragma unroll
        for (int i = 0; i < 8; ++i) {
            float v = accs[t][i];
            if (EPI == 1) v = softplus_fast(v + bv);
            C[(size_t)(mb + i) * N + nn] = v;
        }
    }
}

// ---------------------------------------------------------------------------
// Depthwise causal conv (width 4) + bias + SiLU.  x = xz[..., :DI].
// ---------------------------------------------------------------------------
__global__ __launch_bounds__(256) void conv_silu_kernel(
    const float* __restrict__ xz,
    const float* __restrict__ conv_w,
    const float* __restrict__ conv_b,
    float* __restrict__ xs)
{
    const int idx = blockIdx.x * blockDim.x + threadIdx.x;   // over BL*DI
    if (idx >= BL * DI) return;
    const int d  = idx % DI;
    const int bl = idx / DI;
    const int l  = bl % L_;
    const int b  = bl / L_;

    float acc = conv_b[d];
    #pragma unroll
    for (int k = 0; k < DC; ++k) {
        const int ls = l + k - (DC - 1);
        if (ls >= 0) {
            const float xv = xz[(size_t)(b * L_ + ls) * (2 * DI) + d];
            acc = fmaf(xv, conv_w[d * DC + k], acc);
        }
    }
    xs[idx] = acc * sigmoid_fast(acc);
}

// ---------------------------------------------------------------------------
// Selective scan.  One lane per (b, d) channel; 16 states in registers.
// B_t/C_t (shared across d) staged through LDS in 64-step chunks using the
// CDNA5 async global->LDS path (ASYNCcnt) when available.
// Epilogue fuses y = (y_scan + x*D) * silu(z), written in place over xs.
// ---------------------------------------------------------------------------
#define CHUNK 64
__global__ __launch_bounds__(256) void scan_kernel(
    const float* __restrict__ dbl,
    const float* __restrict__ dt,
    const float* __restrict__ xz,
    const float* __restrict__ A_log,
    const float* __restrict__ Dv,
    float* __restrict__ xs)   // in: silu(conv(x)); out: y
{
    __shared__ float smem[CHUNK][32];          // [l][0:16)=B_t, [16:32)=C_t
    const int b = blockIdx.x / (DI / 256);
    const int d = (blockIdx.x % (DI / 256)) * 256 + threadIdx.x;

    float A[DS];
    #pragma unroll
    for (int n = 0; n < DS; ++n) A[n] = -__expf(A_log[d * DS + n]);
    const float Dd = Dv[d];

    float h[DS];
    #pragma unroll
    for (int n = 0; n < DS; ++n) h[n] = 0.0f;

    for (int l0 = 0; l0 < L_; l0 += CHUNK) {
        __syncthreads();                       // protect prev chunk reads
#ifdef USE_ASYNC_LDS
        #pragma unroll
        for (int t = 0; t < 2; ++t) {
            const int e  = t * 1024 + threadIdx.x * 4;   // 4 floats per async op
            const int lc = e >> 5, j = e & 31;           // j in {0,4,...,28}
            const float* g = dbl + (size_t)(b * L_ + l0 + lc) * 96 + DR + j;
            __builtin_amdgcn_global_load_async_to_lds_b128(
                (as1_v4i*)(uintptr_t)g,
                (as3_v4i*)(uint32_t)(uintptr_t)&smem[lc][j],
                0, 0);
        }
        wait_async_all();
#else
        #pragma unroll
        for (int t = 0; t < (CHUNK * 32) / 256; ++t) {
            const int i  = threadIdx.x + t * 256;
            const int lc = i >> 5, j = i & 31;
            smem[lc][j] = dbl[(size_t)(b * L_ + l0 + lc) * 96 + DR + j];
        }
#endif
        __syncthreads();

        for (int lc = 0; lc < CHUNK; ++lc) {
            const size_t bl  = (size_t)(b * L_ + l0 + lc);
            const float dtv  = dt[bl * DI + d];
            const float xv   = xs[bl * DI + d];
            const float dtx  = dtv * xv;
            float y = 0.0f;
            #pragma unroll
            for (int n = 0; n < DS; ++n) {
                const float dA = __expf(dtv * A[n]);
                h[n] = fmaf(h[n], dA, dtx * smem[lc][n]);
                y    = fmaf(h[n], smem[lc][DS + n], y);
            }
            const float zv = xz[bl * (2 * DI) + DI + d];
            y = fmaf(xv, Dd, y) * (zv * sigmoid_fast(zv));
            xs[bl * DI + d] = y;
        }
    }
}

// ---------------------------------------------------------------------------
extern "C" void kernel_launch(void* const* d_in, const int* in_sizes, int n_in,
                              void* d_out, int out_size, void* d_ws, size_t ws_size,
                              hipStream_t stream)
{
    const float* u          = (const float*)d_in[0];
    const float* in_proj_w  = (const float*)d_in[1];
    const float* conv_w     = (const float*)d_in[2];
    const float* conv_b     = (const float*)d_in[3];
    const float* x_proj_w   = (const float*)d_in[4];
    const float* dt_proj_w  = (const float*)d_in[5];
    const float* dt_proj_b  = (const float*)d_in[6];
    const float* A_log      = (const float*)d_in[7];
    const float* Dv         = (const float*)d_in[8];
    const float* out_proj_w = (const float*)d_in[9];
    float* out = (float*)d_out;

    float* ws  = (float*)d_ws;
    float* xz  = ws;                          // BL * 2*DI   (128 MB)
    float* xs  = xz  + (size_t)BL * 2 * DI;   // BL * DI     ( 64 MB)  x_silu -> y
    float* dbl = xs  + (size_t)BL * DI;       // BL * 96     (  3 MB)
    float* dtb = dbl + (size_t)BL * 96;       // BL * DI     ( 64 MB)

    auto gemm_blocks = [](int M, int N) {
        const long long waves = (long long)((M + 15) / 16) * ((N + 63) / 64);
        return (int)((waves * 32 + 255) / 256);
    };

    // 1) xz = u @ in_proj_w^T                         (8192 x 4096 x 1024)
    gemm_xwt_kernel<0><<<gemm_blocks(BL, 2 * DI), 256, 0, stream>>>(
        u, DM, in_proj_w, nullptr, xz, BL, 2 * DI, DM);

    // 2) xs = silu(causal_conv(x) + conv_b)
    conv_silu_kernel<<<(BL * DI + 255) / 256, 256, 0, stream>>>(
        xz, conv_w, conv_b, xs);

    // 3) dbl = xs @ x_proj_w^T                        (8192 x 96 x 2048)
    gemm_xwt_kernel<0><<<gemm_blocks(BL, DR + 2 * DS), 256, 0, stream>>>(
        xs, DI, x_proj_w, nullptr, dbl, BL, DR + 2 * DS, DI);

    // 4) dt = softplus(dbl[:, :DR] @ dt_proj_w^T + b) (8192 x 2048 x 64, lda=96)
    gemm_xwt_kernel<1><<<gemm_blocks(BL, DI), 256, 0, stream>>>(
        dbl, DR + 2 * DS, dt_proj_w, dt_proj_b, dtb, BL, DI, DR);

    // 5) selective scan + fused (+x*D) * silu(z) -> xs becomes y
    scan_kernel<<<B_ * (DI / 256), 256, 0, stream>>>(
        dbl, dtb, xz, A_log, Dv, xs);

    // 6) out = y @ out_proj_w^T                       (8192 x 1024 x 2048)
    gemm_xwt_kernel<0><<<gemm_blocks(BL, DM), 256, 0, stream>>>(
        xs, DI, out_proj_w, nullptr, out, BL, DM, DI);
}